// GCN_85950885527623
// MI455X (gfx1250) — compile-verified
//
#include <hip/hip_runtime.h>

// GCN 2-layer forward for MI455X (gfx1250, wave32).
// Layer-1 GEMM uses V_WMMA_F32_16X16X4_F32 (exact fp32 matrix pipe).
// Edge aggregation: one wave32 per edge, float2 per lane, f32 atomics to L2.

typedef __attribute__((ext_vector_type(2))) float v2f;
typedef __attribute__((ext_vector_type(8))) float v8f;

#define IN_F 24
#define HID  64

// ---------------- degree / normalization ----------------
__global__ void k_deg_init(float* __restrict__ deg, int n) {
  int i = blockIdx.x * blockDim.x + threadIdx.x;
  if (i < n) deg[i] = 1.0f;                 // self-loop contribution
}

__global__ void k_deg_count(const long long* __restrict__ dst,
                            float* __restrict__ deg, long long nE) {
  long long e = (long long)blockIdx.x * blockDim.x + threadIdx.x;
  if (e < nE) atomicAdd(&deg[dst[e]], 1.0f);
}

__global__ void k_dinv(const float* __restrict__ deg,
                       float* __restrict__ dinv, int n) {
  int i = blockIdx.x * blockDim.x + threadIdx.x;
  if (i < n) {
    float d = deg[i];
    dinv[i] = (d > 0.0f) ? rsqrtf(d) : 0.0f;
  }
}

// ---------------- layer-1 GEMM: h1 = x @ W1 (fp32 WMMA) ----------------
// One wave32 per 16-node tile; 4 accumulators cover all 64 output columns.
__global__ void k_gemm1_wmma(const float* __restrict__ x,
                             const float* __restrict__ W1,
                             float* __restrict__ h1, int nTiles) {
  const int wave = threadIdx.x >> 5;
  const int lane = threadIdx.x & 31;
  const int tile = blockIdx.x * (blockDim.x >> 5) + wave;
  if (tile >= nTiles) return;              // wave-uniform: EXEC stays all-ones

  const int m  = lane & 15;                // A: M index / B: N index
  const int kh = (lane >> 4) << 1;         // K sub-offset: 0 or 2
  const float* xrow = x + (size_t)(tile * 16 + m) * IN_F;

  v8f acc0 = {}, acc1 = {}, acc2 = {}, acc3 = {};
  #pragma unroll
  for (int kk = 0; kk < IN_F; kk += 4) {
    const int k0 = kk + kh;
    v2f a;  a.x = xrow[k0];  a.y = xrow[k0 + 1];
    const float* w0 = W1 + (size_t)k0 * HID + m;        // row k0
    const float* w1 = W1 + (size_t)(k0 + 1) * HID + m;  // row k0+1
    v2f b0;  b0.x = w0[0];   b0.y = w1[0];
    v2f b1;  b1.x = w0[16];  b1.y = w1[16];
    v2f b2;  b2.x = w0[32];  b2.y = w1[32];
    v2f b3;  b3.x = w0[48];  b3.y = w1[48];
    acc0 = __builtin_amdgcn_wmma_f32_16x16x4_f32(false, a, false, b0, (short)0, acc0, false, false);
    acc1 = __builtin_amdgcn_wmma_f32_16x16x4_f32(false, a, false, b1, (short)0, acc1, false, false);
    acc2 = __builtin_amdgcn_wmma_f32_16x16x4_f32(false, a, false, b2, (short)0, acc2, false, false);
    acc3 = __builtin_amdgcn_wmma_f32_16x16x4_f32(false, a, false, b3, (short)0, acc3, false, false);
  }

  // C/D layout: VGPR r, lanes 0-15 -> M=r, N=lane; lanes 16-31 -> M=r+8, N=lane-16
  const int nloc = lane & 15;
  const int mb   = (lane >> 4) * 8;
  float* out = h1 + (size_t)tile * 16 * HID;
  #pragma unroll
  for (int r = 0; r < 8; ++r) {
    float* orow = out + (size_t)(mb + r) * HID + nloc;
    orow[0]  = acc0[r];
    orow[16] = acc1[r];
    orow[32] = acc2[r];
    orow[48] = acc3[r];
  }
}

// Scalar tail for node counts not divisible by 16 (unused for N=100000).
__global__ void k_gemm1_tail(const float* __restrict__ x,
                             const float* __restrict__ W1,
                             float* __restrict__ h1, int start, int n) {
  int idx = blockIdx.x * blockDim.x + threadIdx.x;
  int i = start + idx / HID;
  int f = idx % HID;
  if (i < n) {
    float s = 0.0f;
    #pragma unroll
    for (int k = 0; k < IN_F; ++k) s += x[(size_t)i * IN_F + k] * W1[(size_t)k * HID + f];
    h1[(size_t)i * HID + f] = s;
  }
}

// ---------------- self-loop message init: agg = h1 * dinv^2 ----------------
__global__ void k_selfmsg1(const float* __restrict__ h1,
                           const float* __restrict__ dinv,
                           float* __restrict__ agg, int total) {
  int t = blockIdx.x * blockDim.x + threadIdx.x;
  if (t < total) {
    int i = t >> 6;                         // t / HID
    float w = dinv[i]; w *= w;
    agg[t] = h1[t] * w;
  }
}

// ---------------- edge scatter (layer 1): 64 f32 per edge, 1 wave per edge ----------------
__global__ void k_scatter1(const long long* __restrict__ src,
                           const long long* __restrict__ dst,
                           const float* __restrict__ dinv,
                           const float* __restrict__ h1,
                           float* __restrict__ agg, long long nE) {
  const long long e = (long long)blockIdx.x * (blockDim.x >> 5) + (threadIdx.x >> 5);
  const int lane = threadIdx.x & 31;
  if (e >= nE) return;
  const long long s = src[e];
  const long long d = dst[e];
  const float w = dinv[s] * dinv[d];
  const float2 v = *(const float2*)(h1 + s * HID + lane * 2);   // coalesced 256B/wave
  float* ad = agg + d * HID + lane * 2;
  atomicAdd(ad,     v.x * w);
  atomicAdd(ad + 1, v.y * w);
}

// ---------------- bias + ReLU in place ----------------
__global__ void k_bias_relu(float* __restrict__ agg,
                            const float* __restrict__ b1, int total) {
  int t = blockIdx.x * blockDim.x + threadIdx.x;
  if (t < total) {
    float v = agg[t] + b1[t & (HID - 1)];
    agg[t] = v > 0.0f ? v : 0.0f;
  }
}

// ---------------- layer-2 GEMM (mat-vec, OUT=1): z = hr @ W2 ----------------
__global__ void k_gemm2(const float* __restrict__ hr,
                        const float* __restrict__ W2,
                        float* __restrict__ z, int n) {
  int i = blockIdx.x * blockDim.x + threadIdx.x;
  if (i >= n) return;
  const float4* r = (const float4*)(hr + (size_t)i * HID);
  const float4* w = (const float4*)W2;
  float s = 0.0f;
  #pragma unroll
  for (int f = 0; f < HID / 4; ++f) {
    float4 a = r[f], b = w[f];
    s += a.x * b.x + a.y * b.y + a.z * b.z + a.w * b.w;
  }
  z[i] = s;
}

// ---------------- output init: out = b2 + z * dinv^2 (self-loop) ----------------
__global__ void k_out_init(const float* __restrict__ z,
                           const float* __restrict__ dinv,
                           const float* __restrict__ b2,
                           float* __restrict__ out, int n) {
  int i = blockIdx.x * blockDim.x + threadIdx.x;
  if (i < n) {
    float w = dinv[i];
    out[i] = z[i] * w * w + b2[0];
  }
}

// ---------------- edge scatter (layer 2): 1 f32 per edge ----------------
__global__ void k_scatter2(const long long* __restrict__ src,
                           const long long* __restrict__ dst,
                           const float* __restrict__ dinv,
                           const float* __restrict__ z,
                           float* __restrict__ out, long long nE) {
  long long e = (long long)blockIdx.x * blockDim.x + threadIdx.x;
  if (e < nE) {
    long long s = src[e];
    long long d = dst[e];
    atomicAdd(&out[d], z[s] * dinv[s] * dinv[d]);
  }
}

extern "C" void kernel_launch(void* const* d_in, const int* in_sizes, int n_in,
                              void* d_out, int out_size, void* d_ws, size_t ws_size,
                              hipStream_t stream) {
  const float*     x   = (const float*)d_in[0];
  const long long* ei  = (const long long*)d_in[1];   // int64 [2, E]
  const float*     W1  = (const float*)d_in[2];
  const float*     b1  = (const float*)d_in[3];
  const float*     W2  = (const float*)d_in[4];
  const float*     b2  = (const float*)d_in[5];
  float*           out = (float*)d_out;

  const int n = in_sizes[0] / IN_F;
  const long long E = (long long)in_sizes[1] / 2;
  const long long* src = ei;
  const long long* dst = ei + E;

  // workspace layout (floats, 64-element aligned blocks)
  float* ws = (float*)d_ws;
  auto align64 = [](size_t v) { return (v + 63) & ~(size_t)63; };
  size_t oDeg  = 0;
  size_t oDinv = align64(oDeg  + (size_t)n);
  size_t oZ    = align64(oDinv + (size_t)n);
  size_t oH1   = align64(oZ    + (size_t)n);
  size_t oAgg  = align64(oH1   + (size_t)n * HID);
  float* deg  = ws + oDeg;
  float* dinv = ws + oDinv;
  float* z    = ws + oZ;
  float* h1   = ws + oH1;
  float* agg  = ws + oAgg;

  const int B = 256;                       // 8 wave32s per block
  const int wavesPerBlk = B / 32;
  const int total = n * HID;

  // normalization
  hipLaunchKernelGGL(k_deg_init,  dim3((n + B - 1) / B), dim3(B), 0, stream, deg, n);
  hipLaunchKernelGGL(k_deg_count, dim3((unsigned)((E + B - 1) / B)), dim3(B), 0, stream, dst, deg, E);
  hipLaunchKernelGGL(k_dinv,      dim3((n + B - 1) / B), dim3(B), 0, stream, deg, dinv, n);

  // layer 1
  const int nTiles = n / 16;
  if (nTiles > 0)
    hipLaunchKernelGGL(k_gemm1_wmma, dim3((nTiles + wavesPerBlk - 1) / wavesPerBlk),
                       dim3(B), 0, stream, x, W1, h1, nTiles);
  const int rem = n - nTiles * 16;
  if (rem > 0)
    hipLaunchKernelGGL(k_gemm1_tail, dim3((rem * HID + B - 1) / B), dim3(B), 0, stream,
                       x, W1, h1, nTiles * 16, n);
  hipLaunchKernelGGL(k_selfmsg1, dim3((total + B - 1) / B), dim3(B), 0, stream, h1, dinv, agg, total);
  hipLaunchKernelGGL(k_scatter1, dim3((unsigned)((E + wavesPerBlk - 1) / wavesPerBlk)),
                     dim3(B), 0, stream, src, dst, dinv, h1, agg, E);
  hipLaunchKernelGGL(k_bias_relu, dim3((total + B - 1) / B), dim3(B), 0, stream, agg, b1, total);

  // layer 2
  hipLaunchKernelGGL(k_gemm2,    dim3((n + B - 1) / B), dim3(B), 0, stream, agg, W2, z, n);
  hipLaunchKernelGGL(k_out_init, dim3((n + B - 1) / B), dim3(B), 0, stream, z, dinv, b2, out, n);
  hipLaunchKernelGGL(k_scatter2, dim3((unsigned)((E + B - 1) / B)), dim3(B), 0, stream,
                     src, dst, dinv, z, out, E);
}